// DynamicDWConv_58153857188343
// MI455X (gfx1250) — compile-verified
//
#include <hip/hip_runtime.h>

// ---------------------------------------------------------------------------
// DynamicDWConv for MI455X (gfx1250), fp32 via V_WMMA_F32_16X16X4_F32.
// Conv1/conv2 are implicit GEMM (9 shifted 1x1 GEMMs). K-chunk staging uses
// the Tensor Data Mover (tensor_load_to_lds, TENSORcnt) with DOUBLE BUFFERING:
// wave 0 issues chunk c+1's two tile DMAs into the alternate LDS buffer before
// the workgroup computes chunk c, so the DMA overlaps the 16-WMMA compute
// phase and only one barrier per chunk is needed. conv2 fuses lrelu + global
// average pooling so h2 is never materialized.
// ---------------------------------------------------------------------------

typedef __attribute__((ext_vector_type(2))) float v2f;
typedef __attribute__((ext_vector_type(8))) float v8f;
typedef __attribute__((ext_vector_type(4))) unsigned int u32x4;
typedef __attribute__((ext_vector_type(8))) unsigned int u32x8;

#if defined(__HIP_DEVICE_COMPILE__)
#if !__has_builtin(__builtin_amdgcn_wmma_f32_16x16x4_f32)
#error "device toolchain lacks __builtin_amdgcn_wmma_f32_16x16x4_f32 for gfx1250"
#endif
#endif

#define CCH 256   // channels (Cin == Cout == 256)
#define KK9 9     // 3x3 taps

// ---------------------------------------------------------------------------
// TDM: DMA a 2-D tile (tile_w x tile_h DWORDs) from global into LDS with
// per-row zero-fill past tensor_w and LDS row padding (pitch = tile_w + pad).
//   pad_icode: pad_interval code (5 -> every 64 DWORDs, 6 -> every 128)
//   pad_acode: pad_amount  code (3 -> 4 DWORDs)
// Completion tracked with TENSORcnt (EXEC is ignored for tensor ops).
// ---------------------------------------------------------------------------
__device__ __forceinline__ void tdm_load_2d(const float* gsrc, unsigned lds_addr,
                                            unsigned tile_w, unsigned tile_h,
                                            unsigned tensor_w, unsigned row_stride,
                                            unsigned pad_icode, unsigned pad_acode) {
#if defined(__HIP_DEVICE_COMPILE__)
  unsigned long long ga = (unsigned long long)gsrc;
  u32x4 g0;
  g0[0] = 1u;                                  // count=1, is_restore=0, gather=0
  g0[1] = lds_addr;                            // lds_addr [63:32]
  g0[2] = (unsigned)ga;                        // global_addr [95:64]
  g0[3] = (unsigned)((ga >> 32) & 0x01FFFFFFu) // global_addr [120:96]
          | (2u << 30);                        // type=2 ("image") [127:126]
  u32x8 g1;
  g1[0] = (2u << 16)                           // data_size=2 (4B)
          | (1u << 20)                         // pad_enable
          | (pad_icode << 22)                  // pad_interval
          | (pad_acode << 25);                 // pad_amount
  g1[1] = (tensor_w & 0xFFFFu) << 16;          // tensor_dim0 [79:48] lo16
  g1[2] = (tensor_w >> 16);                    // tensor_dim0 hi16; dim1 lo16=0
  g1[3] = 16u | (tile_w << 16);                // tensor_dim1=0x100000; tile_dim0
  g1[4] = tile_h;                              // tile_dim1; tile_dim2=0
  g1[5] = row_stride;                          // tensor_dim0_stride lo32 (dwords)
  g1[6] = 0u;
  g1[7] = 0u;
  asm volatile("tensor_load_to_lds %0, %1" ::"s"(g0), "s"(g1) : "memory");
#else
  (void)gsrc; (void)lds_addr; (void)tile_w; (void)tile_h;
  (void)tensor_w; (void)row_stride; (void)pad_icode; (void)pad_acode;
#endif
}

// ---------------------------------------------------------------------------
// Repack conv weights [Cout, Cin, 3, 3] -> [off=dy*3+dx][Cin][Cout] so the
// TDM tile rows (cout contiguous) are dense in memory.
// ---------------------------------------------------------------------------
__global__ __launch_bounds__(256) void repack_w_kernel(const float* __restrict__ w,
                                                       float* __restrict__ wr) {
  int i = blockIdx.x * 256 + threadIdx.x;
  if (i < KK9 * CCH * CCH) {
    int off  = i / (CCH * CCH);
    int r    = i - off * (CCH * CCH);
    int cin  = r >> 8;
    int cout = r & 255;
    wr[i] = w[(size_t)cout * (CCH * KK9) + cin * KK9 + off];
  }
}

__global__ void zero_f32_kernel(float* __restrict__ p, int n) {
  int i = blockIdx.x * blockDim.x + threadIdx.x;
  if (i < n) p[i] = 0.0f;
}

// ---------------------------------------------------------------------------
// 3x3 VALID conv + bias + leaky_relu(0.1), optionally fused global-sum pool.
// Block: 256 threads (8 waves) -> 128 couts x 64 output columns of one row.
// Wave: 32x32 subtile -> 4 f32-WMMA accumulators (2 a-frags x 2 b-frags).
// K loop: 144 flattened chunks (9 taps x 16 cin-chunks), TDM double-buffered.
// ---------------------------------------------------------------------------
__global__ __launch_bounds__(256) void conv3x3_wmma_kernel(
    const float* __restrict__ X, const float* __restrict__ Wr,
    const float* __restrict__ Bv, float* __restrict__ Y,
    float* __restrict__ pool, int H, int Win, int Ho, int Wo) {
  __shared__ float sA[2][16][132];  // [buf][cin_local][cout 0..127], pitch 132
  __shared__ float sB[2][16][68];   // [buf][cin_local][ox   0..63],  pitch 68
  __shared__ float sPool[128];

  const int oy   = blockIdx.x;         // output row
  const int co0  = blockIdx.y * 128;   // cout tile base
  const int b    = blockIdx.z;         // batch
  const int tid  = threadIdx.x;
  const int lane = tid & 31;
  const int wave = tid >> 5;
  const int cw   = (wave & 3) * 32;    // cout subtile pair base within 128
  const int pw   = (wave >> 2) * 32;   // pixel subtile pair base (0 or 32)
  const int l16  = lane & 15;
  const int khalf = (lane < 16) ? 0 : 2;  // ISA half-offset K split

  const unsigned ldsA0 = (unsigned)(unsigned long long)(void*)&sA[0][0][0];
  const unsigned ldsA1 = (unsigned)(unsigned long long)(void*)&sA[1][0][0];
  const unsigned ldsB0 = (unsigned)(unsigned long long)(void*)&sB[0][0][0];
  const unsigned ldsB1 = (unsigned)(unsigned long long)(void*)&sB[1][0][0];

  const float* xb = X + (size_t)b * CCH * H * Win;

  // Issue both TDM tile loads for flattened chunk c into buffer `buf`.
  auto issue_chunk = [&](int c, int buf) {
    const int off = c >> 4;            // tap index 0..8
    const int k   = (c & 15) << 4;     // cin chunk base
    const int dy  = off / 3;
    const int dx  = off - dy * 3;
    const float* wsrc = Wr + off * (CCH * CCH) + co0 + k * CCH;
    const float* xsrc = xb + (size_t)k * H * Win + (oy + dy) * Win + dx;
    // weights: 16 rows x 128 dwords, row stride 256 dwords; LDS pitch 132
    tdm_load_2d(wsrc, buf ? ldsA1 : ldsA0, 128u, 16u, 128u, (unsigned)CCH, 6u, 3u);
    // activations: 16 rows x 64 dwords, row stride H*Win dwords; zero-fill
    // past (Win - dx) valid columns; LDS pitch 68
    tdm_load_2d(xsrc, buf ? ldsB1 : ldsB0, 64u, 16u, (unsigned)(Win - dx),
                (unsigned)(H * Win), 5u, 3u);
  };

  v8f a00 = {}, a01 = {}, a10 = {}, a11 = {};

  const int NCH = KK9 * (CCH / 16);  // 144 chunks
  if (wave == 0) {
    issue_chunk(0, 0);
    __builtin_amdgcn_s_wait_tensorcnt(0);
  }
  __syncthreads();

  for (int c = 0; c < NCH; ++c) {
    const int cur = c & 1;
    // Prefetch next chunk into the other buffer; everyone finished reading it
    // one iteration ago (end-of-iteration barrier), so no WAR hazard.
    if (wave == 0 && (c + 1) < NCH) issue_chunk(c + 1, cur ^ 1);
    // ---- 4 WMMA K-steps of 4 cins each; 4 wmma per step per wave ----
#pragma unroll
    for (int j = 0; j < 4; ++j) {
      const int kk = j * 4 + khalf;
      v2f fa0, fa1, fb0, fb1;
      fa0.x = sA[cur][kk][cw + l16];
      fa0.y = sA[cur][kk + 1][cw + l16];
      fa1.x = sA[cur][kk][cw + 16 + l16];
      fa1.y = sA[cur][kk + 1][cw + 16 + l16];
      fb0.x = sB[cur][kk][pw + l16];
      fb0.y = sB[cur][kk + 1][pw + l16];
      fb1.x = sB[cur][kk][pw + 16 + l16];
      fb1.y = sB[cur][kk + 1][pw + 16 + l16];
      a00 = __builtin_amdgcn_wmma_f32_16x16x4_f32(false, fa0, false, fb0,
                                                  (short)0, a00, false, false);
      a01 = __builtin_amdgcn_wmma_f32_16x16x4_f32(false, fa0, false, fb1,
                                                  (short)0, a01, false, false);
      a10 = __builtin_amdgcn_wmma_f32_16x16x4_f32(false, fa1, false, fb0,
                                                  (short)0, a10, false, false);
      a11 = __builtin_amdgcn_wmma_f32_16x16x4_f32(false, fa1, false, fb1,
                                                  (short)0, a11, false, false);
    }
    // Chunk c+1's DMA ran concurrently with the WMMAs above; close it out and
    // publish the buffer to all waves with a single barrier per chunk.
    if (wave == 0) __builtin_amdgcn_s_wait_tensorcnt(0);
    __syncthreads();
  }

  // ---- epilogue: bias + leaky_relu(0.1); masked store; fused pool sums ----
  const int hi  = (lane < 16) ? 0 : 8;
  const int ox0 = pw + l16;
  const int ox1 = pw + 16 + l16;
  if (tid < 128) sPool[tid] = 0.0f;
  __syncthreads();
#pragma unroll
  for (int cs = 0; cs < 2; ++cs) {
#pragma unroll
    for (int r = 0; r < 8; ++r) {
      const int cl   = cw + cs * 16 + r + hi;   // cout within block tile
      const int cout = co0 + cl;
      const float bv = Bv[cout];
      float v0 = (cs == 0 ? a00[r] : a10[r]) + bv;
      float v1 = (cs == 0 ? a01[r] : a11[r]) + bv;
      v0 = (v0 > 0.0f) ? v0 : 0.1f * v0;
      v1 = (v1 > 0.0f) ? v1 : 0.1f * v1;
      if (Y != nullptr) {
        float* yrow = Y + (((size_t)b * CCH + cout) * Ho + oy) * Wo;
        if (ox0 < Wo) yrow[ox0] = v0;
        if (ox1 < Wo) yrow[ox1] = v1;
      }
      if (pool != nullptr) {
        float s = ((ox0 < Wo) ? v0 : 0.0f) + ((ox1 < Wo) ? v1 : 0.0f);
        atomicAdd(&sPool[cl], s);
      }
    }
  }
  if (pool != nullptr) {
    __syncthreads();
    if (tid < 128) atomicAdd(&pool[(size_t)b * CCH + co0 + tid], sPool[tid]);
  }
}

// ---------------------------------------------------------------------------
// kern[b,o] = (1/3600)*sum_c pool[b,c]*w3[o,c] + b3[o]   (M=16,N=2304,K=256)
// One wave per 16-wide N tile; 144 waves total; WMMA f32 16x16x4 K loop.
// ---------------------------------------------------------------------------
__global__ __launch_bounds__(256) void kern_gemm_wmma_kernel(
    const float* __restrict__ p, const float* __restrict__ w3,
    const float* __restrict__ b3, float* __restrict__ kern) {
  const int lane  = threadIdx.x & 31;
  const int wid   = blockIdx.x * 8 + (threadIdx.x >> 5);  // 0..143
  const int o0    = wid * 16;
  const int l16   = lane & 15;
  const int khalf = (lane < 16) ? 0 : 2;
  const float inv = 1.0f / 3600.0f;  // mean over 60x60

  v8f acc = {};
  for (int k = 0; k < CCH; k += 4) {
    const int kk = k + khalf;
    v2f a, bb;
    a.x  = p[l16 * CCH + kk] * inv;       // A[b=l16, c]
    a.y  = p[l16 * CCH + kk + 1] * inv;
    bb.x = w3[(o0 + l16) * CCH + kk];     // B[c, o] = w3[o, c]
    bb.y = w3[(o0 + l16) * CCH + kk + 1];
    acc = __builtin_amdgcn_wmma_f32_16x16x4_f32(false, a, false, bb,
                                                (short)0, acc, false, false);
  }
  const int hi = (lane < 16) ? 0 : 8;
  const float bo = b3[o0 + l16];
#pragma unroll
  for (int r = 0; r < 8; ++r) {
    kern[(size_t)(r + hi) * (KK9 * CCH) + o0 + l16] = acc[r] + bo;
  }
}

// ---------------------------------------------------------------------------
// Per-sample depthwise 3x3 SAME conv + bias. Pure bandwidth (~134 MB).
// One block per (b, c) 64x64 plane; 9 weights broadcast from kern[b, c*9+..].
// ---------------------------------------------------------------------------
__global__ __launch_bounds__(256) void dwconv_kernel(
    const float* __restrict__ x, const float* __restrict__ kern,
    const float* __restrict__ bias, float* __restrict__ y) {
  const int bc = blockIdx.x;         // b*256 + c
  const int b  = bc >> 8;
  const int c  = bc & 255;
  const float* xp = x + (size_t)bc * 4096;
  float* yp = y + (size_t)bc * 4096;
  float w[9];
#pragma unroll
  for (int i = 0; i < 9; ++i) w[i] = kern[(size_t)b * (KK9 * CCH) + c * 9 + i];
  const float bs = bias[c];
  for (int p = threadIdx.x; p < 4096; p += 256) {
    const int row = p >> 6;
    const int col = p & 63;
    float s = bs;
#pragma unroll
    for (int i = 0; i < 3; ++i) {
      const int r = row + i - 1;
      if (r < 0 || r > 63) continue;
#pragma unroll
      for (int j = 0; j < 3; ++j) {
        const int cc = col + j - 1;
        if (cc < 0 || cc > 63) continue;
        s += xp[r * 64 + cc] * w[i * 3 + j];
      }
    }
    yp[p] = s;
  }
}

// ---------------------------------------------------------------------------
extern "C" void kernel_launch(void* const* d_in, const int* in_sizes, int n_in,
                              void* d_out, int out_size, void* d_ws, size_t ws_size,
                              hipStream_t stream) {
  (void)in_sizes; (void)n_in; (void)out_size; (void)ws_size;
  const float* x    = (const float*)d_in[0];  // [16,256,64,64]
  const float* kin  = (const float*)d_in[1];  // [16,256,64,64]
  const float* w1   = (const float*)d_in[2];  // [256,256,3,3]
  const float* b1   = (const float*)d_in[3];  // [256]
  const float* w2   = (const float*)d_in[4];  // [256,256,3,3]
  const float* b2   = (const float*)d_in[5];  // [256]
  const float* w3   = (const float*)d_in[6];  // [2304,256,1,1]
  const float* b3   = (const float*)d_in[7];  // [2304]
  const float* bias = (const float*)d_in[8];  // [256]
  float* out = (float*)d_out;                 // [16,256,64,64]

  // Workspace layout (fp32 elements):
  float* ws   = (float*)d_ws;
  float* h1   = ws;                                       // 16*256*62*62
  float* wr1  = h1 + (size_t)16 * CCH * 62 * 62;          // 9*256*256
  float* wr2  = wr1 + (size_t)KK9 * CCH * CCH;            // 9*256*256
  float* pool = wr2 + (size_t)KK9 * CCH * CCH;            // 16*256 sums
  float* kern = pool + (size_t)16 * CCH;                  // 16*2304

  const int wElems = KK9 * CCH * CCH;  // 589824
  repack_w_kernel<<<(wElems + 255) / 256, 256, 0, stream>>>(w1, wr1);
  repack_w_kernel<<<(wElems + 255) / 256, 256, 0, stream>>>(w2, wr2);
  zero_f32_kernel<<<16, 256, 0, stream>>>(pool, 16 * CCH);

  // conv1: kin [64x64] -> h1 [62x62], lrelu
  dim3 g1(62, 2, 16);
  conv3x3_wmma_kernel<<<g1, 256, 0, stream>>>(kin, wr1, b1, h1, nullptr,
                                              64, 64, 62, 62);
  // conv2: h1 [62x62] -> (lrelu, pooled sums only; h2 never stored)
  dim3 g2(60, 2, 16);
  conv3x3_wmma_kernel<<<g2, 256, 0, stream>>>(h1, wr2, b2, nullptr, pool,
                                              62, 62, 60, 60);
  // kernel predictor: kern = (pool/3600) @ w3^T + b3   [16, 2304]
  kern_gemm_wmma_kernel<<<18, 256, 0, stream>>>(pool, w3, b3, kern);
  // dynamic depthwise conv + bias
  dwconv_kernel<<<16 * CCH, 256, 0, stream>>>(x, kern, bias, out);
}